// ODEFunc_71416716198472
// MI455X (gfx1250) — compile-verified
//
#include <hip/hip_runtime.h>
#include <hip/hip_bf16.h>
#include <cstdint>

// ---------------------------------------------------------------------------
// ODEFunc: f = sigmoid(alpha) * (G @ (G @ x)) - x     (COO SpMM x2, D=64)
// Memory-bound: ~180MB traffic -> ~8us roofline @23.3TB/s. x/y1/out (25.6MB
// each) are L2-resident (192MB L2); edge stream (38.4MB/hop) is streamed with
// NT hint via async global->LDS staging (ASYNCcnt, double buffered). Scatter
// uses no-return global_atomic_add_f32 (L2 atomic units). WMMA structurally
// inapplicable (random COO sparsity).
// ---------------------------------------------------------------------------

#define WAVES_PER_BLOCK 8
#define CHUNK 32   // edges staged per wave per iteration

__device__ __forceinline__ void async_ld_b32_nt(uint32_t lds_off, const void* gptr) {
    uint64_t ga = (uint64_t)(uintptr_t)gptr;
    asm volatile("global_load_async_to_lds_b32 %0, %1, off th:TH_LOAD_NT"
                 :: "v"(lds_off), "v"(ga) : "memory");
}
__device__ __forceinline__ void wait_async_le(int n) {
    if (n == 0) asm volatile("s_wait_asynccnt 0x0" ::: "memory");
    else        asm volatile("s_wait_asynccnt 0x3" ::: "memory");
}

__global__ __launch_bounds__(256)
void zero2_kernel(float* __restrict__ a, float* __restrict__ b, long n4) {
    long i = (long)blockIdx.x * blockDim.x + threadIdx.x;
    long stride = (long)gridDim.x * blockDim.x;
    float4 z = make_float4(0.f, 0.f, 0.f, 0.f);
    for (long k = i; k < n4; k += stride) {
        ((float4*)a)[k] = z;
        ((float4*)b)[k] = z;
    }
}

__global__ __launch_bounds__(256)
void spmm_atomic_kernel(const int*   __restrict__ rows,
                        const int*   __restrict__ cols,
                        const float* __restrict__ vals,
                        const float* __restrict__ src,   // [N,64]
                        float*       __restrict__ dst,   // [N,64] pre-zeroed
                        long E) {
    // packed edge records: {row, col, val_bits, pad} -> one ds_load_b128/edge
    __shared__ int4 s_edge[WAVES_PER_BLOCK][2][CHUNK];

    const int lane = threadIdx.x & 31;
    const int wave = threadIdx.x >> 5;
    const int h    = lane >> 4;            // which edge of the pair
    const uint32_t suboff = (uint32_t)((lane & 15) << 4);  // byte off into row

    const long nchunks = (E + CHUNK - 1) / CHUNK;
    const long waveId  = (long)blockIdx.x * WAVES_PER_BLOCK + wave;
    const long wstride = (long)gridDim.x * WAVES_PER_BLOCK;

    auto issue = [&](long chunk, int b) {
        long e = chunk * CHUNK + lane;
        if (e >= E) e = E - 1;                       // clamp (dup reads are safe)
        uint32_t rec = (uint32_t)(uintptr_t)&s_edge[wave][b][lane];
        async_ld_b32_nt(rec + 0, rows + e);
        async_ld_b32_nt(rec + 4, cols + e);
        async_ld_b32_nt(rec + 8, vals + e);
    };

    const char* srcb = (const char*)src;
    char*       dstb = (char*)dst;

    int buf = 0;
    if (waveId < nchunks) issue(waveId, 0);          // prologue

    for (long c = waveId; c < nchunks; c += wstride) {
        long cn = c + wstride;
        bool havenext = (cn < nchunks);
        if (havenext) issue(cn, buf ^ 1);            // overlap next tile
        wait_async_le(havenext ? 3 : 0);             // current tile resident

        const long base = c * CHUNK;
        if (base + CHUNK <= E) {
            // fast path: whole tile in-bounds, no per-edge guards
            #pragma unroll
            for (int j = 0; j < CHUNK / 2; ++j) {
                const int slot = j * 2 + h;
                const int4 rec = s_edge[wave][buf][slot];   // ds_load_b128 bcast
                const uint32_t row = (uint32_t)rec.x;
                const uint32_t col = (uint32_t)rec.y;
                const float    v   = __int_as_float(rec.z);
                // 32-bit offsets -> saddr + voffset global ops, no 64-bit VALU
                const float4 xv = *(const float4*)(srcb + (col << 8) + suboff);
                float* dp = (float*)(dstb + (row << 8) + suboff);
                atomicAdd(dp + 0, v * xv.x);   // no-return global_atomic_add_f32
                atomicAdd(dp + 1, v * xv.y);
                atomicAdd(dp + 2, v * xv.z);
                atomicAdd(dp + 3, v * xv.w);
            }
        } else {
            // tail tile: executes at most once per wave; keep it small
            #pragma unroll 1
            for (int j = 0; j < CHUNK / 2; ++j) {
                const int slot = j * 2 + h;
                const long e = base + slot;
                const int4 rec = s_edge[wave][buf][slot];
                const uint32_t row = (uint32_t)rec.x;
                const uint32_t col = (uint32_t)rec.y;
                const float    v   = __int_as_float(rec.z);
                if (e < E) {
                    const float4 xv = *(const float4*)(srcb + (col << 8) + suboff);
                    float* dp = (float*)(dstb + (row << 8) + suboff);
                    atomicAdd(dp + 0, v * xv.x);
                    atomicAdd(dp + 1, v * xv.y);
                    atomicAdd(dp + 2, v * xv.z);
                    atomicAdd(dp + 3, v * xv.w);
                }
            }
        }
        buf ^= 1;
    }
}

__global__ __launch_bounds__(256)
void finalize_kernel(const float* __restrict__ x,
                     const float* __restrict__ alpha,
                     float*       __restrict__ out,   // holds G@(G@x); in-place
                     long n4) {
    long i = (long)blockIdx.x * blockDim.x + threadIdx.x;
    long stride = (long)gridDim.x * blockDim.x;
    for (long k = i; k < n4; k += stride) {
        long node = (k * 4) >> 6;                    // D = 64
        float a = alpha[node];
        float g = 1.0f / (1.0f + __expf(-a));
        float4 v  = ((float4*)out)[k];
        float4 xv = ((const float4*)x)[k];
        v.x = g * v.x - xv.x;
        v.y = g * v.y - xv.y;
        v.z = g * v.z - xv.z;
        v.w = g * v.w - xv.w;
        ((float4*)out)[k] = v;
    }
}

extern "C" void kernel_launch(void* const* d_in, const int* in_sizes, int n_in,
                              void* d_out, int out_size, void* d_ws, size_t ws_size,
                              hipStream_t stream) {
    // setup_inputs order: t, x, alpha_train, edge_rows, edge_cols, edge_vals
    const float* x     = (const float*)d_in[1];
    const float* alpha = (const float*)d_in[2];
    const int*   erows = (const int*)  d_in[3];
    const int*   ecols = (const int*)  d_in[4];
    const float* evals = (const float*)d_in[5];

    const long ND = in_sizes[1];   // N * 64
    const long E  = in_sizes[3];

    float* out = (float*)d_out;    // hop-2 accumulator, finalized in place
    float* y1  = (float*)d_ws;     // hop-1 result, ND floats (25.6 MB)

    const long n4 = ND / 4;
    int zgrid = (int)((n4 + 255) / 256); if (zgrid > 2048) zgrid = 2048;
    zero2_kernel<<<zgrid, 256, 0, stream>>>(y1, out, n4);

    spmm_atomic_kernel<<<1024, 256, 0, stream>>>(erows, ecols, evals, x,  y1,  E);
    spmm_atomic_kernel<<<1024, 256, 0, stream>>>(erows, ecols, evals, y1, out, E);

    finalize_kernel<<<zgrid, 256, 0, stream>>>(x, alpha, out, n4);
}